// CausalNSFLayer_16810501997236
// MI455X (gfx1250) — compile-verified
//
#include <hip/hip_runtime.h>
#include <stdint.h>

// ---------------------------------------------------------------------------
// RQS (rational-quadratic spline) flow layer, MI455X / gfx1250.
// N=1048576 rows x 64 dims; dims 0..31 transformed (nodes == arange(32)),
// dims 32..63 pass through. Output: x [N,64] then logdet [N], concatenated.
// ---------------------------------------------------------------------------

#define NROWS      1048576
#define DTOT       64
#define DHALF      32
#define KBINS      40        // spline bins
#define KNOTS      41        // knots per dim (KBINS+1)
#define TBLSTRIDE  41        // odd stride -> conflict-free LDS banking
#define BBOUND     5.0f
#define TILE_ROWS  8
#define TILE_FLOATS (TILE_ROWS * DTOT)   // 512 floats = 2 KB per tile
#define WAVES_PER_BLOCK 8
#define THREADS    (WAVES_PER_BLOCK * 32)
#define NUM_BLOCKS 2048

typedef float v4f __attribute__((ext_vector_type(4)));

__device__ __forceinline__ uint32_t lds_byte_off(const void* p) {
  // generic->LDS: low 32 bits of a flat pointer into LDS are the LDS byte
  // offset (ISA 10.2).
  return (uint32_t)(uintptr_t)p;
}

__global__ __launch_bounds__(THREADS) void rqs_flow_kernel(
    const float* __restrict__ u, const float* __restrict__ w,
    const float* __restrict__ h, const float* __restrict__ d,
    float* __restrict__ out) {
  __shared__ float s_cumw[DHALF * TBLSTRIDE];
  __shared__ float s_cumh[DHALF * TBLSTRIDE];
  __shared__ float s_der [DHALF * TBLSTRIDE];
  __shared__ float s_tile[WAVES_PER_BLOCK][2][TILE_FLOATS];

  const int tid = threadIdx.x;

  // ------------------------------------------------------------------
  // Per-block table build: softmax->cumsum knots, softplus derivatives.
  // 8 threads per dim; shuffle scan inside each aligned 8-lane group.
  // ------------------------------------------------------------------
  {
    const int sub = tid & 7;
    const int dim = tid >> 3;  // 0..31
    for (int which = 0; which < 2; ++which) {
      const float* src = which ? h : w;
      float* dst = which ? s_cumh : s_cumw;
      float e[5];
      float loc = 0.0f;
#pragma unroll
      for (int i = 0; i < 5; ++i) {
        e[i] = __expf(src[dim * KBINS + sub * 5 + i]);
        loc += e[i];
      }
      float v = loc;  // inclusive scan over the 8 subs
#pragma unroll
      for (int off = 1; off < 8; off <<= 1) {
        float o = __shfl_up(v, off, 8);
        if (sub >= off) v += o;
      }
      float total = __shfl(v, 7, 8);
      float run   = v - loc;                 // exclusive prefix
      float scale = (2.0f * BBOUND) / total;
      if (sub == 0) dst[dim * TBLSTRIDE] = -BBOUND;
#pragma unroll
      for (int i = 0; i < 5; ++i) {
        run += e[i];
        dst[dim * TBLSTRIDE + sub * 5 + i + 1] = fmaf(scale, run, -BBOUND);
      }
    }
    if (sub == 0) s_der[dim * TBLSTRIDE]        = 1.0f;
    if (sub == 7) s_der[dim * TBLSTRIDE + KBINS] = 1.0f;
#pragma unroll
    for (int i = 0; i < 5; ++i) {
      int j = sub * 5 + i;               // 0..38 valid interior derivs
      if (j < KBINS - 1) {
        float x = d[dim * (KBINS - 1) + j];
        // numerically stable softplus
        float sp = fmaxf(x, 0.0f) + log1pf(__expf(-fabsf(x)));
        s_der[dim * TBLSTRIDE + j + 1] = sp;
      }
    }
  }
  __syncthreads();

  // ------------------------------------------------------------------
  // Main streaming loop: each wave owns a double-buffered 2 KB LDS tile
  // (8 rows) fed by the CDNA5 async global->LDS engine.
  // ------------------------------------------------------------------
  const int lane = tid & 31;
  const int wv   = tid >> 5;
  float* buf0 = &s_tile[wv][0][0];
  float* buf1 = &s_tile[wv][1][0];
  const uint32_t lb0 = lds_byte_off(buf0);
  const uint32_t lb1 = lds_byte_off(buf1);

  const uint64_t ubase = (uint64_t)(uintptr_t)u;
  float* out_x  = out;
  float* out_ld = out + (size_t)NROWS * DTOT;

  const int numTiles = NROWS / TILE_ROWS;
  const int wid      = blockIdx.x * WAVES_PER_BLOCK + wv;
  const int wstride  = gridDim.x * WAVES_PER_BLOCK;

  auto issue_tile = [&](int t, int parity) {
    uint64_t g = ubase + ((uint64_t)t * TILE_FLOATS + (uint64_t)lane * 4) * 4u;
    uint32_t l = (parity ? lb1 : lb0) + (uint32_t)lane * 16u;
#pragma unroll
    for (int j = 0; j < TILE_FLOATS / 128; ++j) {  // 4 x 512B wave chunks
      asm volatile("global_load_async_to_lds_b128 %0, %1, off"
                   :: "v"(l), "v"(g) : "memory");
      g += 512u;
      l += 512u;
    }
  };

  int parity = 0;
  if (wid < numTiles) issue_tile(wid, 0);

  for (int t = wid; t < numTiles; t += wstride) {
    const int  tn       = t + wstride;
    const bool has_next = tn < numTiles;
    if (has_next) {
      issue_tile(tn, parity ^ 1);
      asm volatile("s_wait_asynccnt 4" ::: "memory");  // current tile landed
    } else {
      asm volatile("s_wait_asynccnt 0" ::: "memory");
    }

    float* tile = parity ? buf1 : buf0;
    const float* kw = &s_cumw[lane * TBLSTRIDE];
    const float* kh = &s_cumh[lane * TBLSTRIDE];
    const float* kd = &s_der [lane * TBLSTRIDE];

#pragma unroll
    for (int r = 0; r < TILE_ROWS; ++r) {
      float x  = tile[r * DTOT + lane];          // dim == lane, banks 0..31
      float xc = fminf(fmaxf(x, -BBOUND), BBOUND);

      // branchless binary search: largest idx with kw[idx] <= xc
      int idx = 0;
#pragma unroll
      for (int s = 32; s > 0; s >>= 1) {
        int cand = idx + s;
        cand = (cand > KBINS) ? KBINS : cand;
        idx  = (kw[cand] <= xc) ? cand : idx;
      }
      idx = (idx > KBINS - 1) ? (KBINS - 1) : idx;

      float cw = kw[idx], cw1 = kw[idx + 1];
      float ch = kh[idx], ch1 = kh[idx + 1];
      float d0 = kd[idx], d1  = kd[idx + 1];
      float bw = cw1 - cw, bh = ch1 - ch;

      float invbw = __builtin_amdgcn_rcpf(bw);
      float delta = bh * invbw;
      float theta = (xc - cw) * invbw;
      float omt   = 1.0f - theta;
      float t1m   = theta * omt;
      float denom = fmaf(fmaf(-2.0f, delta, d0 + d1), t1m, delta);
      float invdn = __builtin_amdgcn_rcpf(denom);
      float num   = bh * fmaf(delta * theta, theta, d0 * t1m);
      float y     = fmaf(num, invdn, ch);
      // logdet = log(delta^2 * A) - 2*log(denom) = log(delta^2*A*invdn^2)
      float A  = fmaf(d1 * theta, theta,
                      fmaf(2.0f * delta, t1m, d0 * omt * omt));
      float L  = (delta * delta) * A * (invdn * invdn);
      float ld = __builtin_amdgcn_logf(L) * 0.6931471805599453f;

      bool inside = fabsf(x) <= BBOUND;
      y  = inside ? y  : x;
      ld = inside ? ld : 0.0f;

      tile[r * DTOT + lane] = y;                 // write back for B128 store

#pragma unroll
      for (int m = 16; m > 0; m >>= 1) ld += __shfl_xor(ld, m, 32);
      if (lane == 0) out_ld[t * TILE_ROWS + r] = ld;
    }

    // streamed B128 store of the whole tile (transformed + passthrough dims)
    {
      uint64_t ob = (uint64_t)(uintptr_t)out_x + (uint64_t)t * TILE_FLOATS * 4u;
#pragma unroll
      for (int j = 0; j < TILE_FLOATS / 128; ++j) {
        v4f v = *(const v4f*)&tile[j * 128 + lane * 4];
        __builtin_nontemporal_store(v, (v4f*)(ob + (uint64_t)(j * 128 + lane * 4) * 4u));
      }
    }
    parity ^= 1;
  }
}

extern "C" void kernel_launch(void* const* d_in, const int* in_sizes, int n_in,
                              void* d_out, int out_size, void* d_ws, size_t ws_size,
                              hipStream_t stream) {
  (void)in_sizes; (void)n_in; (void)out_size; (void)d_ws; (void)ws_size;
  const float* u = (const float*)d_in[0];
  const float* w = (const float*)d_in[1];
  const float* h = (const float*)d_in[2];
  const float* d = (const float*)d_in[3];
  // d_in[4] = nodes (arange(32)) -- identity mapping, folded into the kernel.
  rqs_flow_kernel<<<NUM_BLOCKS, THREADS, 0, stream>>>(u, w, h, d, (float*)d_out);
}